// YoloLoss_44650480009269
// MI455X (gfx1250) — compile-verified
//
#include <hip/hip_runtime.h>
#include <hip/hip_bf16.h>

#define LAMBDA_NOOBJ 0.5f
#define LAMBDA_COORD 5.0f

typedef __attribute__((ext_vector_type(2))) float v2f;
typedef __attribute__((ext_vector_type(8))) float v8f;

#define K1_BLOCKS 2048
#define K1_THREADS 256   // 8 waves (wave32)

// ---------------------------------------------------------------------------
// Deterministic wave32 sum via V_WMMA_F32_16X16X4_F32 with B = ones.
// A layout (32-bit, 16x4): lanes 0-15: v0->K0, v1->K1; lanes 16-31: v0->K2, v1->K3.
// Put partial in v0, zero in v1 => D[m][n] = acc_m + acc_{m+16} (same for all n).
// Lanes 0-15 then hold rows M=0..7 in c[0..7]; lanes 16-31 hold M=8..15.
// One shfl_xor(16) makes every lane carry the full wave total.
// Must be called with full EXEC (all threads reach it, outside divergence).
// ---------------------------------------------------------------------------
__device__ __forceinline__ float wave_sum_wmma(float acc) {
    v2f a; a.x = acc;  a.y = 0.0f;
    v2f b; b.x = 1.0f; b.y = 1.0f;
    v8f c = {};
    c = __builtin_amdgcn_wmma_f32_16x16x4_f32(
            /*neg_a=*/false, a, /*neg_b=*/false, b,
            /*c_mod=*/(short)0, c, /*reuse_a=*/false, /*reuse_b=*/false);
    float s = ((c[0] + c[1]) + (c[2] + c[3])) + ((c[4] + c[5]) + (c[6] + c[7]));
    s += __shfl_xor(s, 16, 32);
    return s;
}

// IoU with the reference's quirk: h == w for both boxes (box[...,3] unused).
__device__ __forceinline__ float iou_xyw(float x1, float y1, float w1,
                                         float x2, float y2, float w2) {
    float w_int = fminf(x1 + w1, x2 + w2) - fmaxf(x1, x2);
    float h_int = fminf(y1 + w1, y2 + w2) - fmaxf(y1, y2);
    float inter = w_int * h_int;
    float uni   = w1 * w1 + w2 * w2 - inter;
    float r     = inter / uni;
    return ((w_int > 0.0f) && (h_int > 0.0f)) ? r : 0.0f;
}

__device__ __forceinline__ float cell_loss(const float* __restrict__ pred,
                                           const float* __restrict__ tgt,
                                           int cell) {
    // Cell payload is 120 bytes => always 8-byte aligned: float2 loads.
    const float2* pp = reinterpret_cast<const float2*>(pred) + (size_t)cell * 15;
    const float2* tp = reinterpret_cast<const float2*>(tgt)  + (size_t)cell * 15;

    float p[30], t[30];
    #pragma unroll
    for (int j = 0; j < 15; ++j) {
        float2 v = pp[j];
        p[2 * j] = v.x; p[2 * j + 1] = v.y;
    }
    #pragma unroll
    for (int j = 0; j < 15; ++j) {
        float2 v = tp[j];
        t[2 * j] = v.x; t[2 * j + 1] = v.y;
    }

    float obj   = (t[4] == 1.0f) ? 1.0f : 0.0f;
    float noobj = 1.0f - obj;

    float iou1 = iou_xyw(p[0], p[1], p[2], t[0], t[1], t[2]);
    float iou2 = iou_xyw(p[5], p[6], p[7], t[0], t[1], t[2]);
    bool  c1   = iou1 > iou2;

    float rx = c1 ? p[0] : p[5];
    float ry = c1 ? p[1] : p[6];
    float rw = c1 ? p[2] : p[7];
    float rh = c1 ? p[3] : p[8];
    float rc = c1 ? p[4] : p[9];   // responsible confidence
    float oc = c1 ? p[9] : p[4];   // other confidence

    float dx = rx - t[0];
    float dy = ry - t[1];
    float xy_loss = LAMBDA_COORD * (dx * dx + dy * dy);

    float sw = sqrtf(fabsf(rw)) - sqrtf(t[2]);
    float sh = sqrtf(fabsf(rh)) - sqrtf(t[3]);
    float wh_loss = LAMBDA_COORD * (sw * sw + sh * sh);

    float od = 1.0f - rc;
    float object_loss = od * od;
    float noobj_resp  = LAMBDA_NOOBJ * oc * oc;                     // * obj
    float noobj_cell  = LAMBDA_NOOBJ * (p[4] * p[4] + p[9] * p[9]); // * noobj

    float class_loss = 0.0f;
    #pragma unroll
    for (int j = 10; j < 30; ++j) {
        float d = p[j] - t[j];
        class_loss += d * d;
    }

    return obj * (xy_loss + wh_loss + object_loss + noobj_resp + class_loss)
         + noobj * noobj_cell;
}

// ---------------------------------------------------------------------------
// Pass 1: grid-stride over cells, one f32 partial per block -> d_ws.
// Fully deterministic: per-thread serial accumulation (fixed order),
// WMMA wave reduction, LDS wave-sum array folded in index order by thread 0.
// ---------------------------------------------------------------------------
__launch_bounds__(K1_THREADS)
__global__ void yolo_loss_pass1(const float* __restrict__ pred,
                                const float* __restrict__ tgt,
                                float* __restrict__ partials,
                                int cells) {
    const int tid    = blockIdx.x * blockDim.x + threadIdx.x;
    const int stride = gridDim.x * blockDim.x;

    float acc = 0.0f;
    for (int cell = tid; cell < cells; cell += stride)
        acc += cell_loss(pred, tgt, cell);

    // All threads reach here (EXEC full) -> WMMA reduction is legal.
    float s = wave_sum_wmma(acc);

    __shared__ float wsum[K1_THREADS / 32];
    const int wave = threadIdx.x >> 5;
    if ((threadIdx.x & 31) == 0) wsum[wave] = s;
    __syncthreads();

    if (threadIdx.x == 0) {
        float b = 0.0f;
        #pragma unroll
        for (int w = 0; w < K1_THREADS / 32; ++w) b += wsum[w];
        partials[blockIdx.x] = b;
    }
}

// ---------------------------------------------------------------------------
// Pass 2: single block folds the block partials in fixed order.
// ---------------------------------------------------------------------------
__launch_bounds__(K1_THREADS)
__global__ void yolo_loss_pass2(const float* __restrict__ partials,
                                float* __restrict__ out,
                                int np) {
    float acc = 0.0f;
    for (int i = threadIdx.x; i < np; i += K1_THREADS)
        acc += partials[i];

    float s = wave_sum_wmma(acc);

    __shared__ float wsum[K1_THREADS / 32];
    const int wave = threadIdx.x >> 5;
    if ((threadIdx.x & 31) == 0) wsum[wave] = s;
    __syncthreads();

    if (threadIdx.x == 0) {
        float total = 0.0f;
        #pragma unroll
        for (int w = 0; w < K1_THREADS / 32; ++w) total += wsum[w];
        out[0] = total;
    }
}

extern "C" void kernel_launch(void* const* d_in, const int* in_sizes, int n_in,
                              void* d_out, int out_size, void* d_ws, size_t ws_size,
                              hipStream_t stream) {
    const float* pred = (const float*)d_in[0];
    const float* tgt  = (const float*)d_in[1];
    float* out        = (float*)d_out;
    float* partials   = (float*)d_ws;   // K1_BLOCKS floats = 8 KB

    // cells = N*S*S ; each cell carries D=30 floats.
    int cells = in_sizes[0] / 30;

    int blocks = (cells + K1_THREADS - 1) / K1_THREADS;
    if (blocks > K1_BLOCKS) blocks = K1_BLOCKS;
    if (blocks < 1) blocks = 1;

    yolo_loss_pass1<<<blocks, K1_THREADS, 0, stream>>>(pred, tgt, partials, cells);
    yolo_loss_pass2<<<1, K1_THREADS, 0, stream>>>(partials, out, blocks);
}